// JetGCN_34342558499213
// MI455X (gfx1250) — compile-verified
//
#include <hip/hip_runtime.h>
#include <hip/hip_bf16.h>

typedef float v2f __attribute__((ext_vector_type(2)));
typedef float v8f __attribute__((ext_vector_type(8)));

#define HID 64
#define OUTD 2

// ---------------- zero init ----------------
__global__ void zero_f32(float* __restrict__ p, int n) {
    int i = blockIdx.x * blockDim.x + threadIdx.x;
    if (i < n) p[i] = 0.0f;
}

// ---------------- encoder: h = BN(relu(x @ W_enc + b_enc)) ----------------
__global__ void encoder_kernel(const float* __restrict__ x, const float* __restrict__ W,
                               const float* __restrict__ b, const float* __restrict__ gamma,
                               const float* __restrict__ beta, const float* __restrict__ mean,
                               const float* __restrict__ var, float* __restrict__ h, int n) {
    int idx = blockIdx.x * blockDim.x + threadIdx.x;
    if (idx >= n * HID) return;
    int i = idx >> 6, j = idx & 63;
    float4 xr = ((const float4*)x)[i];
    float v = xr.x * W[j] + xr.y * W[64 + j] + xr.z * W[128 + j] + xr.w * W[192 + j] + b[j];
    v = fmaxf(v, 0.0f);
    v = (v - mean[j]) * rsqrtf(var[j] + 1e-5f) * gamma[j] + beta[j];
    h[idx] = v;
}

// ---------------- WMMA GEMM: C[M,64] = A[M,64] @ B[64,64] (+bias, relu) ----------------
// f32 WMMA 16x16x4, per-wave 16x16 tile, 16 WMMAs over K=64.
__global__ void gemm64_wmma(const float* __restrict__ A, const float* __restrict__ B,
                            const float* __restrict__ bias, float* __restrict__ C,
                            int M, int do_relu) {
    int wave = (int)((blockIdx.x * blockDim.x + threadIdx.x) >> 5);
    int lane = threadIdx.x & 31;
    int rowtiles = (M + 15) >> 4;
    int ntiles = rowtiles * 4;
    if (wave >= ntiles) return;          // uniform per wave: EXEC stays all-1s
    int rt = wave >> 2, ct = wave & 3;
    int row0 = rt << 4, col0 = ct << 4;
    int m = lane & 15;                    // A: M index / B,C: N index (lane group)
    int half = lane >> 4;
    int kh = half << 1;                   // K sub-offset within 4-chunk: 0 or 2
    int arow_i = row0 + m; if (arow_i >= M) arow_i = M - 1;   // tail clamp (dup rows)
    const float* arow = A + (size_t)arow_i * HID + kh;
    const float* bcol = B + (size_t)kh * HID + col0 + m;
    v8f acc = {};
#pragma unroll
    for (int kk = 0; kk < HID; kk += 4) {
        v2f a = *(const v2f*)(arow + kk);                 // A[row][kk+kh], A[row][kk+kh+1]
        v2f bb;
        bb.x = bcol[(size_t)kk * HID];                    // B[kk+kh][col]
        bb.y = bcol[(size_t)kk * HID + HID];              // B[kk+kh+1][col]
        acc = __builtin_amdgcn_wmma_f32_16x16x4_f32(
            /*neg_a=*/false, a, /*neg_b=*/false, bb,
            /*c_mod=*/(short)0, acc, /*reuse_a=*/false, /*reuse_b=*/false);
    }
#pragma unroll
    for (int r = 0; r < 8; ++r) {
        int row = row0 + r + half * 8;    // C/D layout: VGPR r -> M=r (lanes<16), M=r+8 (lanes>=16)
        int col = col0 + m;
        if (row < M) {
            float v = acc[r];
            if (bias) v += bias[col];
            if (do_relu) v = fmaxf(v, 0.0f);
            C[(size_t)row * HID + col] = v;
        }
    }
}

// ---------------- degree count (dst in-degree) ----------------
__global__ void deg_kernel(const int* __restrict__ dst, float* __restrict__ deg, int E) {
    int e = blockIdx.x * blockDim.x + threadIdx.x;
    if (e < E) atomicAdd(&deg[dst[e]], 1.0f);
}

// ---------------- dis = rsqrt(deg + 1)  (self-loop guarantees deg>=1) ----------------
__global__ void dis_kernel(float* __restrict__ deg, int n) {
    int i = blockIdx.x * blockDim.x + threadIdx.x;
    if (i < n) deg[i] = rsqrtf(deg[i] + 1.0f);
}

// ---------------- edge scatter: agg[dst] += hw[src] * dis[src]*dis[dst] ----------------
// 16 threads per edge, float4 load + 4 f32 atomics. Edges [E, E+n) are self loops.
__global__ void agg_kernel(const int* __restrict__ src, const int* __restrict__ dst,
                           const float* __restrict__ dis, const float* __restrict__ hw,
                           float* __restrict__ agg, int E, int n) {
    int t = blockIdx.x * blockDim.x + threadIdx.x;
    int e = t >> 4, q = t & 15;
    if (e >= E + n) return;
    int s, d;
    if (e < E) { s = src[e]; d = dst[e]; }
    else       { s = e - E; d = s; }
    float nrm = dis[s] * dis[d];
    float4 v = ((const float4*)(hw + (size_t)s * HID))[q];
    float* o = agg + (size_t)d * HID + q * 4;
    atomicAdd(o + 0, v.x * nrm);
    atomicAdd(o + 1, v.y * nrm);
    atomicAdd(o + 2, v.z * nrm);
    atomicAdd(o + 3, v.w * nrm);
}

// ---------------- pool: sums[batch[i]] += relu(agg[i] + b_gcn); cnt[g] += 1 ----------------
__global__ void pool_kernel(const float* __restrict__ agg, const float* __restrict__ b_gcn,
                            const int* __restrict__ batch, float* __restrict__ sums,
                            float* __restrict__ cnt, int n) {
    int t = blockIdx.x * blockDim.x + threadIdx.x;
    int i = t >> 4, q = t & 15;
    if (i >= n) return;
    int g = batch[i];
    float4 v = ((const float4*)(agg + (size_t)i * HID))[q];
    const float* bg = b_gcn + q * 4;
    float* o = sums + (size_t)g * HID + q * 4;
    atomicAdd(o + 0, fmaxf(v.x + bg[0], 0.0f));
    atomicAdd(o + 1, fmaxf(v.y + bg[1], 0.0f));
    atomicAdd(o + 2, fmaxf(v.z + bg[2], 0.0f));
    atomicAdd(o + 3, fmaxf(v.w + bg[3], 0.0f));
    if (q == 0) atomicAdd(&cnt[g], 1.0f);
}

// ---------------- gm = sums / max(cnt, 1) ----------------
__global__ void mean_kernel(const float* __restrict__ sums, const float* __restrict__ cnt,
                            float* __restrict__ gm, int G) {
    int idx = blockIdx.x * blockDim.x + threadIdx.x;
    if (idx >= G * HID) return;
    int g = idx >> 6;
    gm[idx] = sums[idx] / fmaxf(cnt[g], 1.0f);
}

// ---------------- out = act1 @ W_cls2 + b_cls2  (64 -> 2) ----------------
__global__ void out_kernel(const float* __restrict__ act1, const float* __restrict__ W,
                           const float* __restrict__ b, float* __restrict__ out, int G) {
    int t = blockIdx.x * blockDim.x + threadIdx.x;
    if (t >= G * OUTD) return;
    int g = t / OUTD, c = t % OUTD;
    const float* a = act1 + (size_t)g * HID;
    float acc = b[c];
#pragma unroll
    for (int k = 0; k < HID; ++k) acc += a[k] * W[k * OUTD + c];
    out[t] = acc;
}

extern "C" void kernel_launch(void* const* d_in, const int* in_sizes, int n_in,
                              void* d_out, int out_size, void* d_ws, size_t ws_size,
                              hipStream_t stream) {
    const float* x        = (const float*)d_in[0];
    const int*   ei       = (const int*)d_in[1];
    const int*   batch    = (const int*)d_in[2];
    const float* W_enc    = (const float*)d_in[3];
    const float* b_enc    = (const float*)d_in[4];
    const float* bn_gamma = (const float*)d_in[5];
    const float* bn_beta  = (const float*)d_in[6];
    const float* bn_mean  = (const float*)d_in[7];
    const float* bn_var   = (const float*)d_in[8];
    const float* W_gcn    = (const float*)d_in[9];
    const float* b_gcn    = (const float*)d_in[10];
    const float* W_cls1   = (const float*)d_in[11];
    const float* b_cls1   = (const float*)d_in[12];
    const float* W_cls2   = (const float*)d_in[13];
    const float* b_cls2   = (const float*)d_in[14];
    float* out = (float*)d_out;

    const int n = in_sizes[0] / 4;         // 100000 nodes
    const int E = in_sizes[1] / 2;         // 1200000 edges
    const int G = out_size / OUTD;         // 1024 graphs
    const int* src = ei;
    const int* dst = ei + E;

    // workspace layout (floats); agg/deg/sums/cnt contiguous for one zero pass
    float* ws   = (float*)d_ws;
    float* h    = ws;                      // n*64
    float* hw   = h + (size_t)n * HID;     // n*64
    float* agg  = hw + (size_t)n * HID;    // n*64   } zeroed
    float* deg  = agg + (size_t)n * HID;   // n      } zeroed (becomes dis)
    float* sums = deg + n;                 // G*64   } zeroed
    float* cnt  = sums + (size_t)G * HID;  // G      } zeroed
    float* gm   = cnt + G;                 // G*64
    float* act1 = gm + (size_t)G * HID;    // G*64

    const int TB = 256;
    // 1) zero accumulators (agg, deg, sums, cnt are contiguous)
    int nz = n * HID + n + G * HID + G;
    zero_f32<<<(nz + TB - 1) / TB, TB, 0, stream>>>(agg, nz);
    // 2) encoder
    encoder_kernel<<<(n * HID + TB - 1) / TB, TB, 0, stream>>>(
        x, W_enc, b_enc, bn_gamma, bn_beta, bn_mean, bn_var, h, n);
    // 3) hw = h @ W_gcn  (WMMA)
    {
        int waves = ((n + 15) / 16) * 4;
        int thr = waves * 32;
        gemm64_wmma<<<(thr + TB - 1) / TB, TB, 0, stream>>>(h, W_gcn, nullptr, hw, n, 0);
    }
    // 4) degree + normalization
    deg_kernel<<<(E + TB - 1) / TB, TB, 0, stream>>>(dst, deg, E);
    dis_kernel<<<(n + TB - 1) / TB, TB, 0, stream>>>(deg, n);
    // 5) edge + self-loop scatter
    {
        long long thr = (long long)(E + n) * 16;
        agg_kernel<<<(int)((thr + TB - 1) / TB), TB, 0, stream>>>(src, dst, deg, hw, agg, E, n);
    }
    // 6) bias + relu + graph pooling
    {
        long long thr = (long long)n * 16;
        pool_kernel<<<(int)((thr + TB - 1) / TB), TB, 0, stream>>>(agg, b_gcn, batch, sums, cnt, n);
    }
    // 7) graph means
    mean_kernel<<<(G * HID + TB - 1) / TB, TB, 0, stream>>>(sums, cnt, gm, G);
    // 8) classifier layer 1 (WMMA, fused bias+relu)
    {
        int waves = ((G + 15) / 16) * 4;
        int thr = waves * 32;
        gemm64_wmma<<<(thr + TB - 1) / TB, TB, 0, stream>>>(gm, W_cls1, b_cls1, act1, G, 1);
    }
    // 9) final 64->2 projection
    out_kernel<<<(G * OUTD + TB - 1) / TB, TB, 0, stream>>>(act1, W_cls2, b_cls2, out, G);
}